// NodeAttentionPool_74079595921435
// MI455X (gfx1250) — compile-verified
//
#include <hip/hip_runtime.h>
#include <math.h>

typedef __attribute__((ext_vector_type(2))) float v2f;
typedef __attribute__((ext_vector_type(8))) float v8f;

#define C_DIM 128
#define NEG_INF (-3.402823466e38f)

// ---------------------------------------------------------------------------
// Kernel A: logits = x @ W + b via V_WMMA_F32_16X16X4_F32, plus block max.
// One wave (32 lanes) per 16-row tile; 8 waves / block => 128 rows / block.
// A (16x4 f32): lanes 0-15 => M=0..15, VGPR0=K(lo), VGPR1=K(lo+1);
//               lane half selects K pair => per-lane contiguous float2 load.
// B (4x16 f32): W value replicated across all 16 columns => D columns equal;
//               lane 0 holds rows 0..7 in c[0..7], lane 16 holds rows 8..15.
// Requires N % 16 == 0 (true for N = 1,000,000).
// ---------------------------------------------------------------------------
__global__ __launch_bounds__(256) void k_logits(const float* __restrict__ x,
                                                const float* __restrict__ W,
                                                const float* __restrict__ bias,
                                                float* __restrict__ logits,
                                                float* __restrict__ bmax,
                                                int NT) {
  __shared__ __align__(16) float Wsh[C_DIM];
  __shared__ float wmaxsh[8];
  int tid = threadIdx.x;
  if (tid < C_DIM) Wsh[tid] = W[tid];
  __syncthreads();

  int wave = tid >> 5;
  int lane = tid & 31;
  int tile = blockIdx.x * 8 + wave;      // wave-uniform => EXEC all ones inside
  float wave_max = NEG_INF;

  if (tile < NT) {
    int row0 = tile << 4;
    int half = lane >> 4;                                // K-pair selector
    int m = row0 + (lane & 15);                          // row for A operand
    const float* xr = x + (size_t)m * C_DIM + (half << 1);
    const float* wr = Wsh + (half << 1);

    v8f c = {0.f, 0.f, 0.f, 0.f, 0.f, 0.f, 0.f, 0.f};
#pragma unroll
    for (int k = 0; k < C_DIM; k += 4) {
      v2f a  = *(const v2f*)(xr + k);                    // x[m][k+2h : k+2h+1]
      v2f bb = *(const v2f*)(wr + k);                    // W[k+2h : k+2h+1] (bcast cols)
      c = __builtin_amdgcn_wmma_f32_16x16x4_f32(false, a, false, bb,
                                                (short)0, c, false, false);
    }

    float bv = bias[0];
    float r0 = c[0] + bv, r1 = c[1] + bv, r2 = c[2] + bv, r3 = c[3] + bv;
    float r4 = c[4] + bv, r5 = c[5] + bv, r6 = c[6] + bv, r7 = c[7] + bv;

    if ((lane & 15) == 0) {                              // lane 0 & lane 16
      float4* out = (float4*)(logits + row0 + (half << 3));
      out[0] = make_float4(r0, r1, r2, r3);
      out[1] = make_float4(r4, r5, r6, r7);
    }
    float m8 = fmaxf(fmaxf(fmaxf(r0, r1), fmaxf(r2, r3)),
                     fmaxf(fmaxf(r4, r5), fmaxf(r6, r7)));
    wave_max = fmaxf(m8, __shfl_xor(m8, 16, 32));        // combine row halves
  }
  if (lane == 0) wmaxsh[wave] = wave_max;
  __syncthreads();
  if (tid == 0) {
    float bm = wmaxsh[0];
#pragma unroll
    for (int i = 1; i < 8; ++i) bm = fmaxf(bm, wmaxsh[i]);
    bmax[blockIdx.x] = bm;
  }
}

// ---------------------------------------------------------------------------
// Kernel B: single-block max reduction of per-block maxima -> scal[0]
// ---------------------------------------------------------------------------
__global__ __launch_bounds__(1024) void k_reduce_max(const float* __restrict__ v,
                                                     int n, float* __restrict__ scal) {
  __shared__ float sh[1024];
  int tid = threadIdx.x;
  float m = NEG_INF;
  for (int i = tid; i < n; i += 1024) m = fmaxf(m, v[i]);
  sh[tid] = m;
  __syncthreads();
  for (int s = 512; s > 0; s >>= 1) {
    if (tid < s) sh[tid] = fmaxf(sh[tid], sh[tid + s]);
    __syncthreads();
  }
  if (tid == 0) scal[0] = sh[0];
}

// ---------------------------------------------------------------------------
// Kernel C: per-block partial sums of exp(logit - gmax)  (fixed order)
// ---------------------------------------------------------------------------
__global__ __launch_bounds__(256) void k_exp_partial(const float* __restrict__ logits,
                                                     int n, const float* __restrict__ scal,
                                                     float* __restrict__ psum) {
  __shared__ float sh[256];
  int tid = threadIdx.x;
  float gmax = scal[0];
  int base = blockIdx.x * 2048;
  int lim = min(base + 2048, n);
  float s = 0.f;
  for (int i = base + tid; i < lim; i += 256) s += expf(logits[i] - gmax);
  sh[tid] = s;
  __syncthreads();
  for (int st = 128; st > 0; st >>= 1) {
    if (tid < st) sh[tid] += sh[tid + st];
    __syncthreads();
  }
  if (tid == 0) psum[blockIdx.x] = sh[0];
}

// ---------------------------------------------------------------------------
// Kernel D: reduce partial sums -> scal[1] = 1/Z
// ---------------------------------------------------------------------------
__global__ __launch_bounds__(512) void k_reduce_sum(const float* __restrict__ psum,
                                                    int n, float* __restrict__ scal) {
  __shared__ float sh[512];
  int tid = threadIdx.x;
  float s = 0.f;
  for (int i = tid; i < n; i += 512) s += psum[i];
  sh[tid] = s;
  __syncthreads();
  for (int st = 256; st > 0; st >>= 1) {
    if (tid < st) sh[tid] += sh[tid + st];
    __syncthreads();
  }
  if (tid == 0) scal[1] = 1.0f / sh[0];
}

// ---------------------------------------------------------------------------
// Kernel E: alpha[i] = exp(logit[i]-gmax) * invZ
// ---------------------------------------------------------------------------
__global__ __launch_bounds__(256) void k_alpha(const float* __restrict__ logits, int n,
                                               const float* __restrict__ scal,
                                               float* __restrict__ alpha) {
  int i = blockIdx.x * 256 + threadIdx.x;
  if (i < n) alpha[i] = expf(logits[i] - scal[0]) * scal[1];
}

// ---------------------------------------------------------------------------
// Kernel F: pooled[g][c] = sum over segment rows of alpha * x. batch sorted,
// so each block (g, column-quarter) binary-searches its row range. One wave
// per block, 32 columns per wave, deterministic (no atomics), every output
// element written exactly once (empty segments write 0).
// ---------------------------------------------------------------------------
__global__ __launch_bounds__(32) void k_pool(const float* __restrict__ x,
                                             const long long* __restrict__ batch,
                                             const float* __restrict__ logits,
                                             const float* __restrict__ scal,
                                             float* __restrict__ pooled, int N) {
  int g = blockIdx.x;
  int c0 = blockIdx.y << 5;
  int lane = threadIdx.x;

  __shared__ int s_lo, s_hi;
  if (lane == 0) {
    int lo = 0, hi = N;
    while (lo < hi) { int mid = (lo + hi) >> 1;
      if (batch[mid] < (long long)g) lo = mid + 1; else hi = mid; }
    s_lo = lo;
    int lo2 = lo, hi2 = N;
    while (lo2 < hi2) { int mid = (lo2 + hi2) >> 1;
      if (batch[mid] < (long long)(g + 1)) lo2 = mid + 1; else hi2 = mid; }
    s_hi = lo2;
  }
  __syncthreads();
  int start = s_lo, end = s_hi;
  float gmax = scal[0], invZ = scal[1];

  float acc = 0.f;
  for (int rs = start; rs < end; rs += 32) {
    int cnt = min(32, end - rs);
    float sl = 0.f;
    if (lane < cnt) sl = expf(logits[rs + lane] - gmax) * invZ;  // coalesced
    for (int j = 0; j < cnt; ++j) {
      float s = __shfl(sl, j, 32);
      acc += s * x[(size_t)(rs + j) * C_DIM + c0 + lane];        // coalesced 128B
    }
  }
  pooled[(size_t)g * C_DIM + c0 + lane] = acc;
}

// ---------------------------------------------------------------------------
extern "C" void kernel_launch(void* const* d_in, const int* in_sizes, int n_in,
                              void* d_out, int out_size, void* d_ws, size_t ws_size,
                              hipStream_t stream) {
  const float* x = (const float*)d_in[0];
  const float* W = (const float*)d_in[1];
  const float* b = (const float*)d_in[2];
  const long long* batch = (const long long*)d_in[3];   // int64 input

  const int N = in_sizes[3];                 // 1,000,000
  const int G = (out_size - N) / C_DIM;      // 1024

  float* ws = (float*)d_ws;
  float* logits = ws;                        // N floats
  float* bmax   = ws + N;                    // <= 8192 block maxima
  float* psum   = ws + N + 8192;             // <= 2048 partial sums
  float* scal   = ws + N + 8192 + 2048;      // [gmax, invZ]

  float* pooled = (float*)d_out;             // G*C floats
  float* alpha  = pooled + (size_t)G * C_DIM;

  const int NT  = N / 16;                    // 16-row WMMA tiles (N % 16 == 0)
  const int NB1 = (NT + 7) / 8;              // blocks in k_logits (8 waves each)
  const int NB2 = (N + 2047) / 2048;         // blocks in k_exp_partial

  k_logits<<<NB1, 256, 0, stream>>>(x, W, b, logits, bmax, NT);
  k_reduce_max<<<1, 1024, 0, stream>>>(bmax, NB1, scal);
  k_exp_partial<<<NB2, 256, 0, stream>>>(logits, N, scal, psum);
  k_reduce_sum<<<1, 512, 0, stream>>>(psum, NB2, scal);
  k_alpha<<<(N + 255) / 256, 256, 0, stream>>>(logits, N, scal, alpha);
  dim3 gpool(G, C_DIM / 32);
  k_pool<<<gpool, 32, 0, stream>>>(x, batch, logits, scal, pooled, N);
}